// VecInt_22565758173897
// MI455X (gfx1250) — compile-verified
//
#include <hip/hip_runtime.h>

// VecInt (scaling-and-squaring SVF integration), fp32, (2,128,128,128,3).
// Memory-bound gather op: ping-pong buffers are L2-resident (101 MB < 192 MB L2).
// x-adjacent trilinear corners are contiguous (channel-last) -> fetch as one
// 24B read per (z,y) pair: 4 gather addresses per voxel instead of 8.

#define DIM 128
#define VOX_PER_BATCH (DIM * DIM * DIM)          // 2^21
#define FLOATS_PER_BATCH (VOX_PER_BATCH * 3)

__global__ __launch_bounds__(256)
void vecint_scale(const float* __restrict__ in, float* __restrict__ out, int n4) {
    int i = blockIdx.x * blockDim.x + threadIdx.x;
    if (i >= n4) return;
    const float4* in4 = (const float4*)in;
    float4* out4 = (float4*)out;
    float4 v = in4[i];
    const float s = 0.0078125f;                   // 2^-7
    v.x *= s; v.y *= s; v.z *= s; v.w *= s;
    out4[i] = v;
}

__global__ __launch_bounds__(256)
void vecint_compose(const float* __restrict__ src, float* __restrict__ dst, int nvox) {
    int idx = blockIdx.x * blockDim.x + threadIdx.x;
    if (idx >= nvox) return;

    int b   = idx >> 21;                          // voxels per batch = 2^21
    int rem = idx & (VOX_PER_BATCH - 1);
    int z   = rem >> 14;
    int y   = (rem >> 7) & (DIM - 1);
    int x   = rem & (DIM - 1);

    const float* base = src + (size_t)b * FLOATS_PER_BATCH;
    const float* vptr = base + (size_t)rem * 3;

    // gfx1250 prefetch of the streaming read path (global_prefetch_b8)
    __builtin_prefetch(vptr + 3 * 256, 0, 0);

    float v0 = vptr[0];                           // merged global_load_b96
    float v1 = vptr[1];
    float v2 = vptr[2];

    // loc = grid + vec, clamped to [0, 127] (matches reference clip-then-floor)
    float lz = fminf(fmaxf((float)z + v0, 0.0f), 127.0f);
    float ly = fminf(fmaxf((float)y + v1, 0.0f), 127.0f);
    float lx = fminf(fmaxf((float)x + v2, 0.0f), 127.0f);

    float fz = floorf(lz), fy = floorf(ly), fx = floorf(lx);
    int iz0 = (int)fz, iy0 = (int)fy, ix0 = (int)fx;
    float tz = lz - fz, ty = ly - fy, tx = lx - fx;
    float wz0 = 1.0f - tz, wy0 = 1.0f - ty, wx0 = 1.0f - tx;

    int iz1 = min(iz0 + 1, DIM - 1);
    int iy1 = min(iy0 + 1, DIM - 1);

    // Merge the x-pair: both corners live at (..,ixA) and (..,ixA+1).
    // If ix0 == 127 (clamp), both logical corners are voxel 127 = ixA+1:
    // fold weights to (0, wx0+tx) = (0, 1). Identical result, one 24B load.
    int   ixA = min(ix0, DIM - 2);
    float wxA = (ix0 < DIM - 1) ? wx0 : 0.0f;
    float wxB = (ix0 < DIM - 1) ? tx  : 1.0f;

    int z0o = iz0 << 14, z1o = iz1 << 14;         // * 128*128
    int y0o = iy0 << 7,  y1o = iy1 << 7;          // * 128

    float a0 = v0, a1 = v1, a2 = v2;              // out = vec + interp(vec, grid+vec)

#define PAIR(ZO, YO, WZY)                                                  \
    do {                                                                   \
        const float* p = base + (size_t)((ZO) + (YO) + ixA) * 3;           \
        float q0 = p[0], q1 = p[1], q2 = p[2];                             \
        float q3 = p[3], q4 = p[4], q5 = p[5];                             \
        float wzy = (WZY);                                                 \
        a0 = fmaf(wzy, fmaf(wxB, q3, wxA * q0), a0);                       \
        a1 = fmaf(wzy, fmaf(wxB, q4, wxA * q1), a1);                       \
        a2 = fmaf(wzy, fmaf(wxB, q5, wxA * q2), a2);                       \
    } while (0)

    PAIR(z0o, y0o, wz0 * wy0);
    PAIR(z0o, y1o, wz0 * ty);
    PAIR(z1o, y0o, tz  * wy0);
    PAIR(z1o, y1o, tz  * ty);
#undef PAIR

    float* o = dst + (size_t)b * FLOATS_PER_BATCH + (size_t)rem * 3;
    o[0] = a0;                                    // merged global_store_b96
    o[1] = a1;
    o[2] = a2;
}

extern "C" void kernel_launch(void* const* d_in, const int* in_sizes, int n_in,
                              void* d_out, int out_size, void* d_ws, size_t ws_size,
                              hipStream_t stream) {
    const float* flow = (const float*)d_in[0];
    float* out = (float*)d_out;
    float* ws  = (float*)d_ws;

    const int total_floats = in_sizes[0];         // 2*128^3*3 = 12,582,912
    const int nvox = total_floats / 3;            // 4,194,304
    const int n4   = total_floats / 4;            // divisible: 3,145,728

    // vec0 = flow / 2^7  -> ws
    vecint_scale<<<(n4 + 255) / 256, 256, 0, stream>>>(flow, ws, n4);

    // 7 squaring steps, ping-pong ws <-> out. 7 is odd, so the final
    // compose writes into d_out.
    float* src = ws;
    float* dst = out;
    const int blocks = (nvox + 255) / 256;
    for (int it = 0; it < 7; ++it) {
        vecint_compose<<<blocks, 256, 0, stream>>>(src, dst, nvox);
        float* t = src; src = dst; dst = t;
    }
}